// prompt_35837207118425
// MI455X (gfx1250) — compile-verified
//
#include <hip/hip_runtime.h>
#include <hip/hip_bf16.h>
#include <stdint.h>

// Problem constants (match reference)
#define VOCAB_SZ       30522
#define BZ             32
#define SEQ            2048
#define DIM            768
#define NP             16
#define PLACEHOLDER_ID 1

typedef float    v4f __attribute__((ext_vector_type(4)));
typedef uint32_t v4u __attribute__((ext_vector_type(4)));
typedef int      v4i __attribute__((ext_vector_type(4)));
typedef int      v8i __attribute__((ext_vector_type(8)));

// ---------------------------------------------------------------------------
// Kernel 1: one wave per batch row. Stage the row's 2048 token ids (8 KB)
// into LDS via the Tensor Data Mover, then do a wave32 ballot prefix-scan to
// record the sequence index of the i-th placeholder into pos[b*NP + i].
// ---------------------------------------------------------------------------
__global__ __launch_bounds__(32) void scan_positions(const int* __restrict__ ids,
                                                     int* __restrict__ pos)
{
    __shared__ int lds_ids[SEQ];
    const int b    = blockIdx.x;
    const int lane = threadIdx.x & 31;
    const int* rowids = ids + b * SEQ;

#if __has_builtin(__builtin_amdgcn_tensor_load_to_lds)
    {
        // Build a 2D (really 1D: 2048 x 1) tensor descriptor, data_size = 4B.
        uint64_t gaddr   = (uint64_t)(uintptr_t)rowids;
        uint32_t lds_off = (uint32_t)(uintptr_t)
            (__attribute__((address_space(3))) int*)lds_ids;

        v4u g0;
        g0.x = 1u;                                   // count = 1 valid descriptor
        g0.y = lds_off;                              // lds_addr (bytes)
        g0.z = (uint32_t)(gaddr & 0xFFFFFFFFu);      // global_addr[31:0]
        g0.w = (uint32_t)((gaddr >> 32) & 0x1FFFFFFu) | (2u << 30); // addr hi | type=2

        v8i g1;
        g1[0] = 0x20000;                 // workgroup_mask=0, data_size=2 (4 bytes)
        g1[1] = (int)(2048u << 16);      // tensor_dim0 = 2048 (bits 48..79)
        g1[2] = 0x00010000;              // tensor_dim1 = 1    (bits 80..111)
        g1[3] = (int)(2048u << 16);      // tile_dim0  = 2048  (bits 112..127)
        g1[4] = 1;                       // tile_dim1 = 1, tile_dim2 = 0
        g1[5] = 2048;                    // tensor_dim0_stride = 2048
        g1[6] = 0;
        g1[7] = 0;

        v4i z4 = {0, 0, 0, 0};
#if __clang_major__ >= 23
        v8i z8 = {0, 0, 0, 0, 0, 0, 0, 0};
        __builtin_amdgcn_tensor_load_to_lds(g0, g1, z4, z4, z8, 0);
#else
        __builtin_amdgcn_tensor_load_to_lds(g0, g1, z4, z4, 0);
#endif
        __builtin_amdgcn_s_wait_tensorcnt(0);
        asm volatile("" ::: "memory");
    }
#else
    for (int i = lane; i < SEQ; i += 32) lds_ids[i] = rowids[i];
    __builtin_amdgcn_s_waitcnt(0);
#endif

    // wave32 ballot prefix-scan over 64 chunks of 32 ids.
    int base = 0;
    for (int c = 0; c < SEQ / 32; ++c) {
        const int s = (c << 5) + lane;
        const bool m = (lds_ids[s] == PLACEHOLDER_ID);
        const unsigned bal = __builtin_amdgcn_ballot_w32(m);
        const unsigned lt  = (1u << lane) - 1u;
        if (m) {
            const int rank = base + (int)__builtin_popcount(bal & lt);
            if (rank < NP) pos[b * NP + rank] = s;
        }
        base += (int)__builtin_popcount(bal);
    }
}

// ---------------------------------------------------------------------------
// Kernel 2: one wave per output row (65,536 rows of 768 floats).
// Resolve the source row (embedding table or prompt), then stream 6 x 16B per
// lane with non-temporal stores so the 192 MB output does not evict the
// L2-resident 94 MB embedding table.
// ---------------------------------------------------------------------------
__global__ __launch_bounds__(256) void gather_rows(const int*   __restrict__ ids,
                                                   const float* __restrict__ emb,
                                                   const float* __restrict__ prompt,
                                                   const int*   __restrict__ pos,
                                                   float*       __restrict__ out)
{
    const int  lane = threadIdx.x & 31;
    const int  wave = threadIdx.x >> 5;
    const long long row = (long long)blockIdx.x * 8 + wave;   // [0, 65536)
    const int  b = (int)(row >> 11);
    const int  s = (int)(row & 2047);

    const int id = ids[row];

    const float* srcp;
    if (id == PLACEHOLDER_ID) {
        // lanes 0..15 hold the 16 placeholder positions of batch row b;
        // exactly one matches s -> its lane index is the rank.
        const int p = (lane < NP) ? pos[b * NP + lane] : -1;
        const unsigned bal = __builtin_amdgcn_ballot_w32(p == s);
        int rank = (int)__builtin_ctz(bal | 0x80000000u);
        if (rank >= NP) rank = NP - 1;
        srcp = prompt + (size_t)rank * DIM;
    } else {
        srcp = emb + (size_t)id * DIM;
    }

    const v4f* __restrict__ s4 = (const v4f*)srcp;
    v4f*       __restrict__ o4 = (v4f*)(out + row * DIM);

#pragma unroll
    for (int i = 0; i < DIM / 4 / 32; ++i) {          // 6 iterations
        v4f v = s4[i * 32 + lane];                    // global_load_b128 (RT: keep table in L2)
        __builtin_nontemporal_store(v, o4 + i * 32 + lane);  // global_store_b128 th:NT
    }
}

// ---------------------------------------------------------------------------
extern "C" void kernel_launch(void* const* d_in, const int* in_sizes, int n_in,
                              void* d_out, int out_size, void* d_ws, size_t ws_size,
                              hipStream_t stream)
{
    (void)in_sizes; (void)n_in; (void)out_size; (void)ws_size;

    const int*   ids    = (const int*)d_in[0];     // [BZ, SEQ]   int32
    const float* emb    = (const float*)d_in[1];   // [VOCAB, D]  f32
    const float* prompt = (const float*)d_in[2];   // [P, D]      f32
    float*       out    = (float*)d_out;           // [BZ, SEQ, D] f32
    int*         pos    = (int*)d_ws;              // [BZ, NP] scratch (2 KB)

    scan_positions<<<BZ, 32, 0, stream>>>(ids, pos);
    gather_rows<<<(BZ * SEQ) / 8, 256, 0, stream>>>(ids, emb, prompt, pos, out);
}